// CanonicalVQLoss_2439541424694
// MI455X (gfx1250) — compile-verified
//
#include <hip/hip_runtime.h>
#include <math.h>

// ---------------------------------------------------------------------------
// CanonicalVQLoss fused kernel for MI455X (gfx1250, wave32, WMMA).
//   loss = cd_loss(x, y) + emd_loss(x, y) + mean(loss_codebook)
//   x = self_rec_shape_f_ (B,N,3), y = shape (B,N,3); B=8, N=2048.
//
// One workgroup per batch; x/y staged in LDS (32 KB each). Every 16x16
// sq-distance tile comes from V_WMMA_F32_16X16X4_F32 using
//   A row = [x0,x1,x2,1], B row = [-2y0,-2y1,-2y2,|y|^2]  =>  -2x.y + |y|^2
// then + |x|^2 and clamp >= 0. Chamfer mins and the 50-iteration auction
// top-2 scans are fused around the WMMA tiles; auction state is LDS-resident
// (ds_max_u64 bid resolution). Inner loops software-pipeline the LDS B-operand
// (and price) fetch one tile ahead so the s_wait_dscnt before each WMMA is a
// partial-count wait instead of a full drain.
// ---------------------------------------------------------------------------

#define BATCH     8
#define NPTS      2048
#define TIL       16
#define TILES     (NPTS / TIL)     // 128
#define THREADS   512
#define WAVES     (THREADS / 32)   // 16 waves (wave32)
#define EMD_EPS_F 0.005f
#define EMD_ITERS 50

typedef __attribute__((ext_vector_type(2))) float v2f;
typedef __attribute__((ext_vector_type(8))) float v8f;

// A-operand for row-tile mt. 16x4 f32 A layout: lane(0-15)=M with K0/K1 in
// the two VGPRs; lanes 16-31 carry K2/K3.
__device__ __forceinline__ v2f load_a(const float* __restrict__ sax, int mt,
                                      int lane) {
  const int l16 = lane & 15;
  const int k0  = (lane & 16) ? 2 : 0;
  const int m   = (mt << 4) + l16;
  v2f a;
  a.x = sax[m * 4 + k0];
  a.y = sax[m * 4 + k0 + 1];
  return a;
}

// B-operand for column-tile nt (4x16 layout mirrors A).
__device__ __forceinline__ v2f load_b(const float* __restrict__ sby, int nt,
                                      int lane) {
  const int l16 = lane & 15;
  const int k0  = (lane & 16) ? 2 : 0;
  const int n   = (nt << 4) + l16;
  v2f b;
  b.x = sby[n * 4 + k0];
  b.y = sby[n * 4 + k0 + 1];
  return b;
}

// D = A*B + |x|^2, clamped >= 0.  C/D layout: lane l, vgpr v ->
// M = v + (l<16?0:8), N = l%16.
__device__ __forceinline__ v8f wmma_tile(v2f a, v2f b,
                                         const float* __restrict__ xr) {
  v8f c = {};
  // (neg_a, A, neg_b, B, c_mod, C, reuse_a, reuse_b)
  c = __builtin_amdgcn_wmma_f32_16x16x4_f32(false, a, false, b, (short)0, c,
                                            false, false);
#pragma unroll
  for (int v = 0; v < 8; ++v) {
    float d = c[v] + xr[v];
    c[v] = d > 0.0f ? d : 0.0f;
  }
  return c;
}

__global__ __launch_bounds__(THREADS) void emd_cd_kernel(
    const float* __restrict__ X,   // (B,N,3) self_rec_shape_f_
    const float* __restrict__ Y,   // (B,N,3) shape
    float* __restrict__ ws) {      // (B,4) partial sums
  __shared__ float               s_ax[NPTS * 4];     // [x0,x1,x2,1]
  __shared__ float               s_by[NPTS * 4];     // [-2y0,-2y1,-2y2,|y|^2]
  __shared__ float               s_xx[NPTS];         // |x|^2
  __shared__ float               s_price[NPTS];
  __shared__ int                 s_owner[NPTS];
  __shared__ int                 s_assign[NPTS];
  __shared__ unsigned long long  s_colbid[NPTS];     // (bid_bits<<32)|(2047-row)
  __shared__ unsigned int        s_colmin[NPTS];     // float bits (nonneg)
  __shared__ float               s_acc[4];           // cd1, cd2, emd

  const int tid  = threadIdx.x;
  const int lane = tid & 31;
  const int wave = tid >> 5;
  const int l16  = lane & 15;
  const int b    = blockIdx.x;
  const size_t base = (size_t)b * NPTS * 3;

  // ---- stage points into LDS, build WMMA-friendly layouts ----
  for (int i = tid; i < NPTS; i += THREADS) {
    float x0 = X[base + i * 3 + 0];
    float x1 = X[base + i * 3 + 1];
    float x2 = X[base + i * 3 + 2];
    s_ax[i * 4 + 0] = x0;
    s_ax[i * 4 + 1] = x1;
    s_ax[i * 4 + 2] = x2;
    s_ax[i * 4 + 3] = 1.0f;
    s_xx[i] = x0 * x0 + x1 * x1 + x2 * x2;
    float y0 = Y[base + i * 3 + 0];
    float y1 = Y[base + i * 3 + 1];
    float y2 = Y[base + i * 3 + 2];
    s_by[i * 4 + 0] = -2.0f * y0;
    s_by[i * 4 + 1] = -2.0f * y1;
    s_by[i * 4 + 2] = -2.0f * y2;
    s_by[i * 4 + 3] = y0 * y0 + y1 * y1 + y2 * y2;
    s_price[i]  = 0.0f;
    s_owner[i]  = -1;
    s_colmin[i] = 0x7F800000u;  // +inf bits
  }
  if (tid < 4) s_acc[tid] = 0.0f;
  __syncthreads();

  // ---- chamfer: row mins (registers) + column mins (ds_min_u32) ----
  float sum1 = 0.0f;
  for (int mt = wave; mt < TILES; mt += WAVES) {
    const int mbase = (mt << 4) + ((lane & 16) ? 8 : 0);
    const v2f a = load_a(s_ax, mt, lane);
    float xr[8];
#pragma unroll
    for (int v = 0; v < 8; ++v) xr[v] = s_xx[mbase + v];
    float pm[8];
#pragma unroll
    for (int v = 0; v < 8; ++v) pm[v] = __builtin_huge_valf();

    v2f bc = load_b(s_by, 0, lane);              // pipelined B operand
#pragma unroll 2
    for (int nt = 0; nt < TILES; ++nt) {
      v2f bn = load_b(s_by, (nt + 1) & (TILES - 1), lane);  // prefetch next
      v8f d = wmma_tile(a, bc, xr);
      float cm = d[0];
#pragma unroll
      for (int v = 1; v < 8; ++v) cm = fminf(cm, d[v]);
      atomicMin(&s_colmin[(nt << 4) + l16], __float_as_uint(cm));
#pragma unroll
      for (int v = 0; v < 8; ++v) pm[v] = fminf(pm[v], d[v]);
      bc = bn;
    }
#pragma unroll
    for (int off = 1; off <= 8; off <<= 1) {
#pragma unroll
      for (int v = 0; v < 8; ++v)
        pm[v] = fminf(pm[v], __shfl_xor(pm[v], off, 32));
    }
    if (l16 == 0) {
#pragma unroll
      for (int v = 0; v < 8; ++v) sum1 += sqrtf(pm[v]);
    }
  }
  if (l16 == 0) atomicAdd(&s_acc[0], sum1);
  __syncthreads();
  {
    float sum2 = 0.0f;
    for (int c = tid; c < NPTS; c += THREADS)
      sum2 += sqrtf(__uint_as_float(s_colmin[c]));
    atomicAdd(&s_acc[1], sum2);
  }

  // ---- auction EMD: 50 iterations, all state in LDS ----
  for (int it = 0; it < EMD_ITERS; ++it) {
    __syncthreads();
    for (int i = tid; i < NPTS; i += THREADS) {
      s_assign[i] = -1;
      s_colbid[i] = 0ULL;
    }
    __syncthreads();
    for (int i = tid; i < NPTS; i += THREADS) {
      int o = s_owner[i];
      if (o >= 0) s_assign[o] = i;
    }
    __syncthreads();

    for (int mt = wave; mt < TILES; mt += WAVES) {
      const int mbase = (mt << 4) + ((lane & 16) ? 8 : 0);
      const v2f a = load_a(s_ax, mt, lane);
      float xr[8];
#pragma unroll
      for (int v = 0; v < 8; ++v) xr[v] = s_xx[mbase + v];
      float v1[8], v2[8];
      int j1[8];
#pragma unroll
      for (int v = 0; v < 8; ++v) {
        v1[v] = -__builtin_huge_valf();
        v2[v] = -__builtin_huge_valf();
        j1[v] = 0;
      }

      v2f bc = load_b(s_by, 0, lane);     // pipelined B operand
      float prc = s_price[l16];           // pipelined price
#pragma unroll 2
      for (int nt = 0; nt < TILES; ++nt) {
        const int ntn = (nt + 1) & (TILES - 1);
        v2f bn = load_b(s_by, ntn, lane);             // prefetch next tile
        float prn = s_price[(ntn << 4) + l16];
        v8f d = wmma_tile(a, bc, xr);
        const int n = (nt << 4) + l16;
#pragma unroll
        for (int v = 0; v < 8; ++v) {
          float val = -d[v] - prc;
          if (val > v1[v]) {
            v2[v] = v1[v];
            v1[v] = val;
            j1[v] = n;
          } else if (val > v2[v]) {
            v2[v] = val;
          }
        }
        bc = bn;
        prc = prn;
      }
      // cross-lane top-2 merge within each 16-lane half
#pragma unroll
      for (int off = 1; off <= 8; off <<= 1) {
#pragma unroll
        for (int v = 0; v < 8; ++v) {
          float ov1 = __shfl_xor(v1[v], off, 32);
          float ov2 = __shfl_xor(v2[v], off, 32);
          int   oj  = __shfl_xor(j1[v], off, 32);
          if (ov1 > v1[v]) {
            v2[v] = fmaxf(v1[v], ov2);
            v1[v] = ov1;
            j1[v] = oj;
          } else {
            v2[v] = fmaxf(v2[v], ov1);
          }
        }
      }
      if (l16 == 0) {
#pragma unroll
        for (int v = 0; v < 8; ++v) {
          int r = mbase + v;
          if (s_assign[r] < 0) {
            int j = j1[v];
            float bid = s_price[j] + (v1[v] - v2[v]) + EMD_EPS_F;
            unsigned long long key =
                ((unsigned long long)__float_as_uint(bid) << 32) |
                (unsigned long long)(unsigned)(NPTS - 1 - r);
            atomicMax(&s_colbid[j], key);  // ds_max_u64
          }
        }
      }
    }
    __syncthreads();
    for (int c = tid; c < NPTS; c += THREADS) {
      unsigned long long kb = s_colbid[c];
      if (kb != 0ULL) {
        s_price[c] = __uint_as_float((unsigned)(kb >> 32));
        s_owner[c] = (NPTS - 1) - (int)(kb & 0xFFFFFFFFULL);
      }
    }
  }
  __syncthreads();

  // ---- final assignment (+ argmin fallback for unmatched rows) ----
  for (int i = tid; i < NPTS; i += THREADS) s_assign[i] = -1;
  __syncthreads();
  for (int i = tid; i < NPTS; i += THREADS) {
    int o = s_owner[i];
    if (o >= 0) s_assign[o] = i;
  }
  __syncthreads();
  for (int r = tid; r < NPTS; r += THREADS) {
    if (s_assign[r] < 0) {
      float a0 = s_ax[r * 4 + 0], a1 = s_ax[r * 4 + 1], a2 = s_ax[r * 4 + 2];
      float xxr = s_xx[r];
      float best = __builtin_huge_valf();
      int bi = 0;
      for (int c = 0; c < NPTS; ++c) {
        float d = xxr + a0 * s_by[c * 4 + 0] + a1 * s_by[c * 4 + 1] +
                  a2 * s_by[c * 4 + 2] + s_by[c * 4 + 3];
        if (d < best) {
          best = d;
          bi = c;
        }
      }
      s_assign[r] = bi;
    }
  }
  __syncthreads();

  // ---- matched EMD distances ----
  float esum = 0.0f;
  for (int r = tid; r < NPTS; r += THREADS) {
    int c = s_assign[r];
    // y = -0.5 * s_by[0..2]  =>  x - y = a + 0.5*by
    float dx = s_ax[r * 4 + 0] + 0.5f * s_by[c * 4 + 0];
    float dy = s_ax[r * 4 + 1] + 0.5f * s_by[c * 4 + 1];
    float dz = s_ax[r * 4 + 2] + 0.5f * s_by[c * 4 + 2];
    esum += sqrtf(dx * dx + dy * dy + dz * dz);
  }
  atomicAdd(&s_acc[2], esum);
  __syncthreads();

  if (tid == 0) {
    ws[b * 4 + 0] = s_acc[0];
    ws[b * 4 + 1] = s_acc[1];
    ws[b * 4 + 2] = s_acc[2];
  }
}

__global__ void finalize_kernel(const float* __restrict__ ws,
                                const float* __restrict__ cb,
                                float* __restrict__ out) {
  if (threadIdx.x == 0 && blockIdx.x == 0) {
    float s = 0.0f;
    for (int b = 0; b < BATCH; ++b)
      s += ws[b * 4 + 0] + ws[b * 4 + 1] + ws[b * 4 + 2];
    // mean over (B,N) for cd1 + cd2 + emd, plus mean(loss_codebook) (1 elem)
    out[0] = s / (float)(BATCH * NPTS) + cb[0];
  }
}

extern "C" void kernel_launch(void* const* d_in, const int* in_sizes, int n_in,
                              void* d_out, int out_size, void* d_ws,
                              size_t ws_size, hipStream_t stream) {
  (void)in_sizes; (void)n_in; (void)out_size; (void)ws_size;
  const float* cb = (const float*)d_in[0];  // loss_codebook (1,)
  const float* Y  = (const float*)d_in[2];  // shape (B,N,3)
  const float* X  = (const float*)d_in[6];  // self_rec_shape_f_ (B,N,3)
  float* ws = (float*)d_ws;                 // B*4 floats of partials
  emd_cd_kernel<<<dim3(BATCH), dim3(THREADS), 0, stream>>>(X, Y, ws);
  finalize_kernel<<<dim3(1), dim3(32), 0, stream>>>(ws, cb, (float*)d_out);
}